// seismic_farword_28046136442963
// MI455X (gfx1250) — compile-verified
//
#include <hip/hip_runtime.h>

// ---------------- problem constants (match reference) ----------------
#define DXc   10.0f
#define DTc   0.004f
#define NTc   500
#define PMLc  20
#define NSHOT 4
#define NRECc 300
#define NXc   300
#define NZc   100
#define Hc    (NXc + 2 * PMLc)   // 340 (dim0)
#define Wc    (NZc + 2 * PMLc)   // 140 (dim1, contiguous)
#define HWc   (Hc * Wc)          // 47600

#define C0 (-205.0f / 72.0f)
#define C1 (8.0f / 5.0f)
#define C2 (-1.0f / 5.0f)
#define C3 (8.0f / 315.0f)
#define C4 (-1.0f / 560.0f)

// ---------------- tiling: 256 threads = 8 wave32 per block -----------
#define TZ   32                  // contiguous (z) extent per tile
#define TX   8                   // row (x) extent per tile
#define HALO 4
#define LTZ  (TZ + 2 * HALO)     // 40
#define LTX  (TX + 2 * HALO)     // 16
#define LSZ  (LTX * LTZ)         // 640 floats = 2.5 KB LDS
#define TILES_X ((Hc + TX - 1) / TX)   // 43
#define TILES_Z ((Wc + TZ - 1) / TZ)   // 5
#define NTILES  (NSHOT * TILES_X * TILES_Z) // 860

typedef __attribute__((address_space(1))) int glb_int;
typedef __attribute__((address_space(3))) int lds_int;

// CDNA5 async copy engine: global -> LDS, tracked by ASYNCcnt.
__device__ __forceinline__ void async_g2l_f32(float* lds_dst, const float* g_src) {
#if defined(__has_builtin) && __has_builtin(__builtin_amdgcn_global_load_async_to_lds_b32)
    __builtin_amdgcn_global_load_async_to_lds_b32(
        (glb_int*)g_src, (lds_int*)lds_dst, 0, 0);
#else
    *lds_dst = *g_src;  // fallback: synchronous copy through VGPR
#endif
}

__device__ __forceinline__ void wait_async0() {
#if defined(__has_builtin) && __has_builtin(__builtin_amdgcn_s_wait_asynccnt)
    __builtin_amdgcn_s_wait_asynccnt(0);
#else
    asm volatile("s_wait_asynccnt 0" ::: "memory");
#endif
}

// ---------------- one leapfrog step (+ deferred receiver gather) -----
// parity at step t: prev = buf[t&1], cur = buf[1-(t&1)], un -> buf[t&1]
// Launch t records step t-1 (its un == cur buffer, untouched this launch).
__global__ void seis_step_kernel(const float* __restrict__ coef,
                                 const float* __restrict__ am,
                                 const float* __restrict__ ap,
                                 float* __restrict__ ubuf,
                                 const float* __restrict__ samp,
                                 const int* __restrict__ sloc,
                                 const int* __restrict__ rloc,
                                 float* __restrict__ out,
                                 int t) {
    // ---- last block: receiver gather for previous step ----
    if ((int)blockIdx.x == (int)gridDim.x - 1) {
        if (t > 0) {
            const float* cur = ubuf + (size_t)(1 - (t & 1)) * (NSHOT * HWc);
            for (int i = threadIdx.x; i < NSHOT * NRECc; i += blockDim.x) {
                const int shot = i / NRECc;
                const int r0 = rloc[i * 2 + 0] + PMLc;
                const int r1 = rloc[i * 2 + 1] + PMLc;
                out[(size_t)i * NTc + (t - 1)] = cur[(size_t)shot * HWc + r0 * Wc + r1];
            }
        }
        return;
    }
    if (t >= NTc) return;  // final launch is record-only

    // ---- tile decode ----
    const int tileId = blockIdx.x;
    const int shot   = tileId / (TILES_X * TILES_Z);
    const int rem    = tileId % (TILES_X * TILES_Z);
    const int tzb    = (rem % TILES_Z) * TZ;
    const int txb    = (rem / TILES_Z) * TX;

    const float* cur  = ubuf + (size_t)(1 - (t & 1)) * (NSHOT * HWc) + (size_t)shot * HWc;
    const float* prev = ubuf + (size_t)(t & 1)       * (NSHOT * HWc) + (size_t)shot * HWc;
    float*       nxt  = ubuf + (size_t)(t & 1)       * (NSHOT * HWc) + (size_t)shot * HWc;

    // ---- this thread's output point; start independent global loads
    //      EARLY so they overlap the async copy engine ----
    const int lz = threadIdx.x & (TZ - 1);
    const int lx = threadIdx.x >> 5;   // TZ == 32
    const int gx = txb + lx;
    const int gz = tzb + lz;
    const bool active = (gx < Hc) && (gz < Wc);
    const int gi = active ? (gx * Wc + gz) : 0;
    float c = 0.0f, a = 1.0f, amv = 0.0f, pv = 0.0f;
    if (active) {
        c   = coef[gi];
        a   = ap[gi];
        amv = am[gi];
        pv  = prev[gi];
    }
    const int s0 = sloc[shot * 2 + 0] + PMLc;   // scalar (uniform) loads
    const int s1 = sloc[shot * 2 + 1] + PMLc;

    // ---- stage (TX+8)x(TZ+8) tile of `cur` into LDS via async copies ----
    __shared__ float tile_s[LSZ];
    for (int i = threadIdx.x; i < LSZ; i += blockDim.x) {
        const int hx = i / LTZ;
        const int hz = i % LTZ;
        const int ux = txb - HALO + hx;
        const int uz = tzb - HALO + hz;
        if (ux >= 0 && ux < Hc && uz >= 0 && uz < Wc)
            async_g2l_f32(&tile_s[i], &cur[ux * Wc + uz]);
        else
            tile_s[i] = 0.0f;  // matches jnp.pad zero boundary
    }
    wait_async0();
    __syncthreads();

    // ---- 8th-order Laplacian + leapfrog update ----
    if (active) {
        const int ci = (lx + HALO) * LTZ + (lz + HALO);
        const float u = tile_s[ci];
        float lap = 2.0f * C0 * u;
        lap = lap + C1 * (((tile_s[ci - 1 * LTZ] + tile_s[ci + 1 * LTZ]) + tile_s[ci - 1]) + tile_s[ci + 1]);
        lap = lap + C2 * (((tile_s[ci - 2 * LTZ] + tile_s[ci + 2 * LTZ]) + tile_s[ci - 2]) + tile_s[ci + 2]);
        lap = lap + C3 * (((tile_s[ci - 3 * LTZ] + tile_s[ci + 3 * LTZ]) + tile_s[ci - 3]) + tile_s[ci + 3]);
        lap = lap + C4 * (((tile_s[ci - 4 * LTZ] + tile_s[ci + 4 * LTZ]) + tile_s[ci - 4]) + tile_s[ci + 4]);

        float un = (2.0f * u - amv * pv + c * lap) / a;

        // single source per shot: s_loc shape (NSHOT, 1, 2)
        if (gx == s0 && gz == s1)
            un += samp[(size_t)shot * NTc + t] * (c / a);

        nxt[gi] = un;
    }
}

// ---------------- setup: vmax reduce, PML coeffs, zero fields --------
__global__ void seis_setup_kernel(const float* __restrict__ vmap,
                                  float* __restrict__ coef,
                                  float* __restrict__ am,
                                  float* __restrict__ ap,
                                  float* __restrict__ ubuf) {
    __shared__ float red[1024];
    const int tid = threadIdx.x;

    float m = 0.0f;
    for (int i = tid; i < NXc * NZc; i += 1024) m = fmaxf(m, vmap[i]);
    red[tid] = m;
    __syncthreads();
    for (int s = 512; s > 0; s >>= 1) {
        if (tid < s) red[tid] = fmaxf(red[tid], red[tid + s]);
        __syncthreads();
    }
    const float vmax = red[0];
    // smax = 3*vmax*ln(1000)/(2*PML*DX)
    const float smax = 3.0f * vmax * 6.907755278982137f / (2.0f * (float)PMLc * DXc);

    for (int i = tid; i < HWc; i += 1024) {
        const int x = i / Wc;
        const int z = i % Wc;
        float dx = fmaxf(fmaxf((float)(PMLc - x), (float)(x - (Hc - 1 - PMLc))), 0.0f) * (1.0f / (float)PMLc);
        float dz = fmaxf(fmaxf((float)(PMLc - z), (float)(z - (Wc - 1 - PMLc))), 0.0f) * (1.0f / (float)PMLc);
        const float sig = smax * dx * dx + smax * dz * dz;
        ap[i] = 1.0f + 0.5f * DTc * sig;
        am[i] = 1.0f - 0.5f * DTc * sig;
        // edge-replicated velocity pad
        int cx = x - PMLc; cx = cx < 0 ? 0 : (cx > NXc - 1 ? NXc - 1 : cx);
        int cz = z - PMLc; cz = cz < 0 ? 0 : (cz > NZc - 1 ? NZc - 1 : cz);
        const float v = vmap[cx * NZc + cz];
        const float r = DTc / DXc;
        coef[i] = r * r * v * v;
    }
    for (int i = tid; i < 2 * NSHOT * HWc; i += 1024) ubuf[i] = 0.0f;
}

// ---------------- host side ----------------
extern "C" void kernel_launch(void* const* d_in, const int* in_sizes, int n_in,
                              void* d_out, int out_size, void* d_ws, size_t ws_size,
                              hipStream_t stream) {
    const float* vmap = (const float*)d_in[0];  // (300,100) f32
    const float* samp = (const float*)d_in[1];  // (4,1,500) f32
    const int*   sloc = (const int*)d_in[2];    // (4,1,2) i32
    const int*   rloc = (const int*)d_in[3];    // (4,300,2) i32
    float* out = (float*)d_out;                 // (4,300,500) f32
    float* ws  = (float*)d_ws;

    float* coef = ws;               // HWc
    float* am   = ws + HWc;         // HWc
    float* ap   = ws + 2 * HWc;     // HWc
    float* ubuf = ws + 3 * HWc;     // 2 * NSHOT * HWc (ping-pong fields)

    seis_setup_kernel<<<1, 1024, 0, stream>>>(vmap, coef, am, ap, ubuf);

    for (int t = 0; t < NTc; ++t)
        seis_step_kernel<<<NTILES + 1, 256, 0, stream>>>(coef, am, ap, ubuf,
                                                         samp, sloc, rloc, out, t);
    // final record-only launch (records step NT-1)
    seis_step_kernel<<<1, 256, 0, stream>>>(coef, am, ap, ubuf,
                                            samp, sloc, rloc, out, NTc);
}